// RSSM_88029649699138
// MI455X (gfx1250) — compile-verified
//
#include <hip/hip_runtime.h>
#include <math.h>

typedef __attribute__((ext_vector_type(2))) float v2f;
typedef __attribute__((ext_vector_type(8))) float v8f;

#define B_TOT     16384
#define T_STEPS   64
#define XH_STRIDE 74   // 72 K-values (s:0..31 | a:32..39 | h:40..71), padded, even
#define WT_STRIDE 34   // 32 K-values padded, even
#define NWAVES    4
#define THREADS   (NWAVES * 32)

// ---- CDNA5 hardware transcendentals (v_tanh_f32 / v_exp_f32 / v_log_f32) ----
#if __has_builtin(__builtin_amdgcn_tanhf)
__device__ __forceinline__ float fast_tanh(float x) { return __builtin_amdgcn_tanhf(x); }
#elif __has_builtin(__builtin_amdgcn_tanh_f32)
__device__ __forceinline__ float fast_tanh(float x) { return __builtin_amdgcn_tanh_f32(x); }
#else
__device__ __forceinline__ float fast_tanh(float x) { return tanhf(x); }
#endif

__device__ __forceinline__ float fast_exp(float x) { return __expf(x); }  // v_exp_f32 path
__device__ __forceinline__ float fast_elu(float x) {
    return (x > 0.0f) ? x : (fast_exp(x) - 1.0f);
}
__device__ __forceinline__ float fast_softplus(float x) {
    // stable: max(x,0) + log(1 + exp(-|x|)); exp arg in (-inf,0], 1+e in (1,2]
    return fmaxf(x, 0.0f) + __logf(1.0f + fast_exp(-fabsf(x)));
}

__global__ __launch_bounds__(THREADS) void rssm_rollout_wmma(
    const float* __restrict__ s0, const float* __restrict__ h0,
    const float* __restrict__ actions,
    const float* __restrict__ W_ih, const float* __restrict__ W_hh,
    const float* __restrict__ b_ih, const float* __restrict__ b_hh,
    const float* __restrict__ fc1_w, const float* __restrict__ fc1_b,
    const float* __restrict__ mean_w, const float* __restrict__ mean_b,
    const float* __restrict__ std_w, const float* __restrict__ std_b,
    float* __restrict__ out)
{
    // ---- LDS: shared weights (transposed: [n][k]) + per-wave activation tiles
    __shared__ float sW1T [32 * XH_STRIDE];          // [n][k], k<40: W_ih, k>=40: W_hh
    __shared__ float sF1T [32 * WT_STRIDE];
    __shared__ float sMT  [32 * WT_STRIDE];
    __shared__ float sST  [32 * WT_STRIDE];
    __shared__ float sBC  [32];                      // b_ih + b_hh
    __shared__ float sBF  [32];
    __shared__ float sBM  [32];
    __shared__ float sBS  [32];
    __shared__ float sXH  [NWAVES][16 * XH_STRIDE];  // per-wave [row][k] : s|a|h
    __shared__ float sHID [NWAVES][16 * WT_STRIDE];  // per-wave [row][k] : elu(fc1)

    const int tid = threadIdx.x;

    // ---- cooperative weight load (transpose into [n][k] with padded stride)
    for (int i = tid; i < 32 * 72; i += THREADS) {
        int n = i / 72, k = i % 72;
        float w = (k < 40) ? W_ih[n * 40 + k] : W_hh[n * 32 + (k - 40)];
        sW1T[n * XH_STRIDE + k] = w;
    }
    for (int i = tid; i < 32 * 32; i += THREADS) {
        int n = i >> 5, k = i & 31;
        sF1T[n * WT_STRIDE + k] = fc1_w[i];
        sMT [n * WT_STRIDE + k] = mean_w[i];
        sST [n * WT_STRIDE + k] = std_w[i];
    }
    if (tid < 32) {
        sBC[tid] = b_ih[tid] + b_hh[tid];
        sBF[tid] = fc1_b[tid];
        sBM[tid] = mean_b[tid];
        sBS[tid] = std_b[tid];
    }

    const int wave  = tid >> 5;          // wave32
    const int lane  = tid & 31;
    const int lhalf = lane >> 4;         // 0: lanes 0-15, 1: lanes 16-31
    const int lmod  = lane & 15;
    const int tile  = blockIdx.x * NWAVES + wave;
    const int rbase = tile * 16;         // first batch row of this wave's tile

    float* xh  = &sXH [wave][0];
    float* hid = &sHID[wave][0];

    // ---- init per-wave tile: s0 -> cols 0..31, h0 -> cols 40..71 (row-major)
    for (int r = 0; r < 16; ++r) {
        xh[r * XH_STRIDE + lane]      = s0[(size_t)(rbase + r) * 32 + lane];
        xh[r * XH_STRIDE + 40 + lane] = h0[(size_t)(rbase + r) * 32 + lane];
    }
    __syncthreads();

    // per-lane bias broadcast values (C/D layout: N = lmod (+16 for hi tile))
    const float bc_lo = sBC[lmod],      bc_hi = sBC[16 + lmod];
    const float bf_lo = sBF[lmod],      bf_hi = sBF[16 + lmod];
    const float bm_lo = sBM[lmod],      bm_hi = sBM[16 + lmod];
    const float bs_lo = sBS[lmod],      bs_hi = sBS[16 + lmod];

    // A-operand base pointers (lane L<16: row L, K=4k+{0,1}; lane>=16: K=4k+{2,3})
    const float* aX  = &xh [lmod * XH_STRIDE +      2 * lhalf]; // layer1 A (K=0..71)
    const float* aH  = &xh [lmod * XH_STRIDE + 40 + 2 * lhalf]; // fc1 A (h, K=0..31)
    const float* aG  = &hid[lmod * WT_STRIDE +      2 * lhalf]; // mean/std A
    // B-operand base pointers (N = lmod / 16+lmod, K pair per lane half)
    const float* w1L = &sW1T[ lmod       * XH_STRIDE + 2 * lhalf];
    const float* w1H = &sW1T[(16 + lmod) * XH_STRIDE + 2 * lhalf];
    const float* f1L = &sF1T[ lmod       * WT_STRIDE + 2 * lhalf];
    const float* f1H = &sF1T[(16 + lmod) * WT_STRIDE + 2 * lhalf];
    const float* mL  = &sMT [ lmod       * WT_STRIDE + 2 * lhalf];
    const float* mH  = &sMT [(16 + lmod) * WT_STRIDE + 2 * lhalf];
    const float* stL = &sST [ lmod       * WT_STRIDE + 2 * lhalf];
    const float* stH = &sST [(16 + lmod) * WT_STRIDE + 2 * lhalf];

    for (int t = 0; t < T_STEPS; ++t) {
        // ---- load this step's actions into xh cols 32..39 (4 floats per lane)
        {
            const int r  = lmod;
            const int c4 = lhalf * 4;
            const float* ap = actions + ((size_t)(rbase + r) * T_STEPS + t) * 8 + c4;
            float4 a4 = *(const float4*)ap;
            float* dst = &xh[r * XH_STRIDE + 32 + c4];
            ((v2f*)dst)[0] = (v2f){a4.x, a4.y};
            ((v2f*)dst)[1] = (v2f){a4.z, a4.w};
            if (t + 1 < T_STEPS) __builtin_prefetch(ap + 8, 0, 0); // global_prefetch_b8
        }

        // ---- layer 1: pre = [s|a]@W_ih^T + h@W_hh^T + (b_ih+b_hh), K = 72
        v8f pre_lo, pre_hi;
#pragma unroll
        for (int r = 0; r < 8; ++r) { pre_lo[r] = bc_lo; pre_hi[r] = bc_hi; }
#pragma unroll
        for (int kc = 0; kc < 18; ++kc) {
            v2f a   = *(const v2f*)(aX  + kc * 4);
            v2f blo = *(const v2f*)(w1L + kc * 4);
            v2f bhi = *(const v2f*)(w1H + kc * 4);
            pre_lo = __builtin_amdgcn_wmma_f32_16x16x4_f32(false, a, false, blo, (short)0, pre_lo, false, false);
            pre_hi = __builtin_amdgcn_wmma_f32_16x16x4_f32(false, a, false, bhi, (short)0, pre_hi, false, false);
        }

        // ---- h = tanh(pre); scatter C/D layout back into xh cols 40..71
#pragma unroll
        for (int r = 0; r < 8; ++r) {
            float hl = fast_tanh(pre_lo[r]);
            float hh = fast_tanh(pre_hi[r]);
            const int m = r + 8 * lhalf;     // C/D: lanes>=16 hold rows M+8
            xh[m * XH_STRIDE + 40 + lmod] = hl;
            xh[m * XH_STRIDE + 56 + lmod] = hh;
        }

        // ---- fc1: hid = elu(h @ fc1_w^T + fc1_b), K = 32
        v8f g_lo, g_hi;
#pragma unroll
        for (int r = 0; r < 8; ++r) { g_lo[r] = bf_lo; g_hi[r] = bf_hi; }
#pragma unroll
        for (int kc = 0; kc < 8; ++kc) {
            v2f a   = *(const v2f*)(aH  + kc * 4);
            v2f blo = *(const v2f*)(f1L + kc * 4);
            v2f bhi = *(const v2f*)(f1H + kc * 4);
            g_lo = __builtin_amdgcn_wmma_f32_16x16x4_f32(false, a, false, blo, (short)0, g_lo, false, false);
            g_hi = __builtin_amdgcn_wmma_f32_16x16x4_f32(false, a, false, bhi, (short)0, g_hi, false, false);
        }
#pragma unroll
        for (int r = 0; r < 8; ++r) {
            float el = fast_elu(g_lo[r]);
            float eh = fast_elu(g_hi[r]);
            const int m = r + 8 * lhalf;
            hid[m * WT_STRIDE + lmod]      = el;
            hid[m * WT_STRIDE + 16 + lmod] = eh;
        }

        // ---- mean = hid @ mean_w^T + mean_b, K = 32
        v8f mn_lo, mn_hi;
#pragma unroll
        for (int r = 0; r < 8; ++r) { mn_lo[r] = bm_lo; mn_hi[r] = bm_hi; }
#pragma unroll
        for (int kc = 0; kc < 8; ++kc) {
            v2f a   = *(const v2f*)(aG + kc * 4);
            v2f blo = *(const v2f*)(mL + kc * 4);
            v2f bhi = *(const v2f*)(mH + kc * 4);
            mn_lo = __builtin_amdgcn_wmma_f32_16x16x4_f32(false, a, false, blo, (short)0, mn_lo, false, false);
            mn_hi = __builtin_amdgcn_wmma_f32_16x16x4_f32(false, a, false, bhi, (short)0, mn_hi, false, false);
        }

        // ---- std_pre = hid @ std_w^T + std_b, K = 32
        v8f sd_lo, sd_hi;
#pragma unroll
        for (int r = 0; r < 8; ++r) { sd_lo[r] = bs_lo; sd_hi[r] = bs_hi; }
#pragma unroll
        for (int kc = 0; kc < 8; ++kc) {
            v2f a   = *(const v2f*)(aG  + kc * 4);
            v2f blo = *(const v2f*)(stL + kc * 4);
            v2f bhi = *(const v2f*)(stH + kc * 4);
            sd_lo = __builtin_amdgcn_wmma_f32_16x16x4_f32(false, a, false, blo, (short)0, sd_lo, false, false);
            sd_hi = __builtin_amdgcn_wmma_f32_16x16x4_f32(false, a, false, bhi, (short)0, sd_hi, false, false);
        }

        // ---- emit outputs [t][b][mean(32)|std(32)]; feed mean back as s_{t}
#pragma unroll
        for (int r = 0; r < 8; ++r) {
            const int m = r + 8 * lhalf;
            float* orow = out + ((size_t)t * B_TOT + rbase + m) * 64;
            float ml = mn_lo[r], mh = mn_hi[r];
            orow[lmod]      = ml;
            orow[16 + lmod] = mh;
            xh[m * XH_STRIDE + lmod]      = ml;   // s_t = mean (mean_only rollout)
            xh[m * XH_STRIDE + 16 + lmod] = mh;
            float sl = fast_softplus(sd_lo[r]) + 1e-5f;
            float sh = fast_softplus(sd_hi[r]) + 1e-5f;
            orow[32 + lmod] = sl;
            orow[48 + lmod] = sh;
        }
    }
}

extern "C" void kernel_launch(void* const* d_in, const int* in_sizes, int n_in,
                              void* d_out, int out_size, void* d_ws, size_t ws_size,
                              hipStream_t stream) {
    const float* s0      = (const float*)d_in[0];
    const float* h0      = (const float*)d_in[1];
    const float* actions = (const float*)d_in[2];
    const float* W_ih    = (const float*)d_in[3];
    const float* W_hh    = (const float*)d_in[4];
    const float* b_ih    = (const float*)d_in[5];
    const float* b_hh    = (const float*)d_in[6];
    const float* fc1_w   = (const float*)d_in[7];
    const float* fc1_b   = (const float*)d_in[8];
    const float* mean_w  = (const float*)d_in[9];
    const float* mean_b  = (const float*)d_in[10];
    const float* std_w   = (const float*)d_in[11];
    const float* std_b   = (const float*)d_in[12];
    // d_in[13] = horizon_length (fixed 64 in this harness)
    float* out = (float*)d_out;

    dim3 grid(B_TOT / (NWAVES * 16));   // 256 blocks
    dim3 block(THREADS);                // 128 threads = 4 waves
    rssm_rollout_wmma<<<grid, block, 0, stream>>>(
        s0, h0, actions, W_ih, W_hh, b_ih, b_hh,
        fc1_w, fc1_b, mean_w, mean_b, std_w, std_b, out);
}